// LlamaMoETopKRouter_55138790146427
// MI455X (gfx1250) — compile-verified
//
#include <hip/hip_runtime.h>
#include <cmath>

typedef __attribute__((ext_vector_type(2))) float v2f;
typedef __attribute__((ext_vector_type(4))) float v4f;
typedef __attribute__((ext_vector_type(8))) float v8f;

#define NUM_TOK    16384
#define H          4096
#define NEXP       8
#define CHUNK      1024
#define WSTRIDE    (CHUNK + 4)   // 1028 dwords: stride%64==4 -> conflict-free b64 reads
#define TOK_PER_WG 128

__global__ __launch_bounds__(256)
void moe_router_wmma(const float* __restrict__ hs,
                     const float* __restrict__ rw,
                     float* __restrict__ out) {
  __shared__ float sW[9 * WSTRIDE];      // 8 expert rows + 1 zero row (experts 8..15 pad)
  __shared__ float sLog[8 * 16 * 17];    // per-wave 16x16 logits tile, row stride 17

  const int tid  = threadIdx.x;
  const int lane = tid & 31;
  const int wv   = tid >> 5;             // wave id 0..7
  const int hi   = lane >> 4;            // K offset 0 / 2 (A & B layouts, ISA 7.12.2)
  const int lo   = lane & 15;            // M (A) / N (B,C)

  const int tok = blockIdx.x * TOK_PER_WG + wv * 16 + lo;
  const float* aBase = hs + (size_t)tok * H + 2 * hi;
  const int   bBase  = ((lo < NEXP) ? lo : NEXP) * WSTRIDE + 2 * hi;

  v8f c = {0.f, 0.f, 0.f, 0.f, 0.f, 0.f, 0.f, 0.f};

  for (int kc = 0; kc < H / CHUNK; ++kc) {
    __syncthreads();
    // Stage router_weight K-chunk into LDS: 8 rows x 1024 floats, padded stride.
    #pragma unroll
    for (int i = 0; i < 8; ++i) {
      int f  = i * 256 + tid;            // float4 index, 0..2047
      int e  = f >> 8;                   // expert row (256 float4 per row-chunk)
      int cc = (f & 255) << 2;           // column in floats
      v4f w = *(const v4f*)(rw + (size_t)e * H + kc * CHUNK + cc);
      *(v4f*)&sW[e * WSTRIDE + cc] = w;
    }
    if (kc == 0) {                        // zero pad-row once; never overwritten
      for (int i = tid; i < WSTRIDE; i += 256) sW[NEXP * WSTRIDE + i] = 0.f;
    }
    __syncthreads();

    const float* ap = aBase + kc * CHUNK;
    #pragma unroll 4
    for (int kk = 0; kk < CHUNK; kk += 4) {
      v2f a = *(const v2f*)(ap + kk);            // global_load_b64 (A frag)
      v2f b = *(const v2f*)&sW[bBase + kk];      // ds_load_b64     (B frag)
      c = __builtin_amdgcn_wmma_f32_16x16x4_f32(false, a, false, b,
                                                (short)0, c, false, false);
    }
  }

  // Spill C tile: VGPR v holds (M = v + 8*hi, N = lo)
  #pragma unroll
  for (int v = 0; v < 8; ++v)
    sLog[wv * 272 + (v + 8 * hi) * 17 + lo] = c[v];
  __syncthreads();

  // 128 threads: one token each -> top-2 + renormalized weights + logits out
  if (tid < TOK_PER_WG) {
    const float* lp = &sLog[(tid >> 4) * 272 + (tid & 15) * 17];
    float lg[NEXP];
    #pragma unroll
    for (int e = 0; e < NEXP; ++e) lg[e] = lp[e];

    int i1 = 0; float m1 = lg[0];
    #pragma unroll
    for (int e = 1; e < NEXP; ++e) { if (lg[e] > m1) { m1 = lg[e]; i1 = e; } }
    int i2 = (i1 == 0) ? 1 : 0; float m2 = lg[i2];
    #pragma unroll
    for (int e = 0; e < NEXP; ++e) {
      if (e != i1 && lg[e] > m2) { m2 = lg[e]; i2 = e; }
    }

    // renormalized top-2 softmax weights (stable; independent of other experts)
    float d   = expf(m2 - m1);
    float inv = 1.0f / (1.0f + d);

    int gt = blockIdx.x * TOK_PER_WG + tid;
    float* outW = out;                       // [16384][2]
    float* outI = out + 2 * NUM_TOK;         // [16384][2] (indices as float)
    float* outL = out + 4 * NUM_TOK;         // [16384][8]
    outW[gt * 2 + 0] = inv;
    outW[gt * 2 + 1] = d * inv;
    outI[gt * 2 + 0] = (float)i1;
    outI[gt * 2 + 1] = (float)i2;
    #pragma unroll
    for (int e = 0; e < NEXP; ++e) outL[gt * NEXP + e] = lg[e];
  }
}

extern "C" void kernel_launch(void* const* d_in, const int* in_sizes, int n_in,
                              void* d_out, int out_size, void* d_ws, size_t ws_size,
                              hipStream_t stream) {
  const float* hs = (const float*)d_in[0];   // [4,4096,4096] f32
  const float* rw = (const float*)d_in[1];   // [8,4096] f32
  float* out = (float*)d_out;                // weights | indices | logits
  (void)in_sizes; (void)n_in; (void)out_size; (void)d_ws; (void)ws_size;

  moe_router_wmma<<<dim3(NUM_TOK / TOK_PER_WG), dim3(256), 0, stream>>>(hs, rw, out);
}